// SimpleContrastiveLoss_72541997629822
// MI455X (gfx1250) — compile-verified
//
#include <hip/hip_runtime.h>
#include <hip/hip_bf16.h>
#include <math.h>

typedef __attribute__((ext_vector_type(2))) float v2f;
typedef __attribute__((ext_vector_type(8))) float v8f;
typedef __attribute__((__vector_size__(16))) int v4i_vs;   // builtin's pointee type

#define NROWS     8192
#define DIMS      256
#define INV_T     (1.0f / 0.07f)
#define MWAVES    8                   // waves per workgroup (wave32)
#define MBLOCK    (MWAVES * 16)       // 128 i-rows per workgroup
#define JSLICES   16
#define JSPAN     (NROWS / JSLICES)   // 512 j-columns per slice
#define LDSTRIDE  260                 // 256 + 4 pad -> conflict-free ds_load_b64
#define AS_FLOATS (MBLOCK * LDSTRIDE)         // 33280
#define BS_FLOATS (16 * LDSTRIDE)             // 4160 per buffer
#define LDS_BYTES ((AS_FLOATS + 2 * BS_FLOATS) * 4)   // 166400 B

// ---------------------------------------------------------------------------
// CDNA5 async global->LDS helpers (guarded; fallback = sync copy)
// ---------------------------------------------------------------------------
#if __has_builtin(__builtin_amdgcn_global_load_async_to_lds_b128)
#define HAS_ASYNC_LDS 1
#else
#define HAS_ASYNC_LDS 0
#endif

static __device__ __forceinline__ void copy16_to_lds(const float* g, float* l) {
#if HAS_ASYNC_LDS
    __builtin_amdgcn_global_load_async_to_lds_b128(
        (__attribute__((address_space(1))) v4i_vs*)(void*)(const_cast<float*>(g)),
        (__attribute__((address_space(3))) v4i_vs*)(void*)l, 0, 0);
#else
    *(float4*)l = *(const float4*)g;
#endif
}

static __device__ __forceinline__ void wait_async0() {
#if HAS_ASYNC_LDS
#if __has_builtin(__builtin_amdgcn_s_wait_asynccnt)
    __builtin_amdgcn_s_wait_asynccnt(0);
#else
    asm volatile("s_wait_asynccnt 0" ::: "memory");
#endif
#endif
}

// ---------------------------------------------------------------------------
// Kernel 1: per-row 1/max(||z_i||,eps), all-ones attribute flag, zero accums.
// ---------------------------------------------------------------------------
__global__ void __launch_bounds__(256)
prep_kernel(const float* __restrict__ z, const float* __restrict__ attr,
            float* __restrict__ invnorm, float* __restrict__ flagf,
            float* __restrict__ all_acc, float* __restrict__ pos_acc) {
    const int lane = threadIdx.x & 31;
    const int wv   = threadIdx.x >> 5;
    const int row  = blockIdx.x * (blockDim.x >> 5) + wv;
    if (row >= NROWS) return;

    const float4* zr = (const float4*)(z + (size_t)row * DIMS);
    float4 a = zr[lane * 2 + 0];
    float4 b = zr[lane * 2 + 1];
    float s = a.x*a.x + a.y*a.y + a.z*a.z + a.w*a.w
            + b.x*b.x + b.y*b.y + b.z*b.z + b.w*b.w;
#pragma unroll
    for (int m = 16; m >= 1; m >>= 1) s += __shfl_xor(s, m, 32);

    if (lane == 0) {
        float n = fmaxf(sqrtf(s), 1e-12f);
        invnorm[row] = 1.0f / n;
        const float* ar = attr + (size_t)row * 4;
        // binary attrs: dot(a_i,a_j)==4  <=>  both rows all-ones
        flagf[row] = (ar[0] + ar[1] + ar[2] + ar[3] == 4.0f) ? 1.0f : 0.0f;
        all_acc[row] = 0.0f;
        pos_acc[row] = 0.0f;
    }
}

// ---------------------------------------------------------------------------
// Kernel 2: LDS-staged WMMA GEMM with fused exp + masked row reductions.
//   As: 128 i-rows staged once per workgroup (async -> LDS)
//   Bs: 16-row j-tiles, double buffered, prefetched ahead of compute
// ---------------------------------------------------------------------------
__global__ void __launch_bounds__(256)
gemm_exp_kernel(const float* __restrict__ z,
                const float* __restrict__ invnorm,
                const float* __restrict__ flagf,
                float* __restrict__ all_acc, float* __restrict__ pos_acc) {
    extern __shared__ float smem[];
    float* As = smem;                         // [MBLOCK][LDSTRIDE]
    float* Bs = smem + AS_FLOATS;             // [2][16][LDSTRIDE]

    const int tid   = threadIdx.x;
    const int lane  = tid & 31;
    const int wave  = tid >> 5;
    const int half  = lane >> 4;              // K sub-pair select
    const int l15   = lane & 15;
    const int iBase = blockIdx.x * MBLOCK;
    const int i0    = iBase + wave * 16;
    const int jBase = blockIdx.y * JSPAN;

    // ---- stage A block (128 rows x 1KB) + first B tile, asynchronously ----
    // 16B chunks: 64 per row. A: 8192 chunks / 256 thr = 32 each.
    {
#pragma unroll 4
        for (int it = 0; it < 32; ++it) {
            int idx = it * 256 + tid;
            int row = idx >> 6, c16 = (idx & 63) * 4;
            copy16_to_lds(z + (size_t)(iBase + row) * DIMS + c16,
                          As + row * LDSTRIDE + c16);
        }
#pragma unroll
        for (int it = 0; it < 4; ++it) {      // B tile 0: 1024 chunks
            int idx = it * 256 + tid;
            int row = idx >> 6, c16 = (idx & 63) * 4;
            copy16_to_lds(z + (size_t)(jBase + row) * DIMS + c16,
                          Bs + row * LDSTRIDE + c16);
        }
    }
    wait_async0();
    __syncthreads();

    // per-lane i-row scales; C layout: (m,n) = (r + 8*half, l15)
    float invNi[8];
#pragma unroll
    for (int r = 0; r < 8; ++r) invNi[r] = invnorm[i0 + r + 8 * half];

    const float* aLds = As + (wave * 16 + l15) * LDSTRIDE + 2 * half;

    v8f allSum = {};
    v8f posSum = {};
    int buf = 0;

    for (int jt = 0; jt < JSPAN; jt += 16) {
        const int j0 = jBase + jt;

        // prefetch next B tile into the other buffer
        if (jt + 16 < JSPAN) {
            float* dst = Bs + (buf ^ 1) * BS_FLOATS;
#pragma unroll
            for (int it = 0; it < 4; ++it) {
                int idx = it * 256 + tid;
                int row = idx >> 6, c16 = (idx & 63) * 4;
                copy16_to_lds(z + (size_t)(j0 + 16 + row) * DIMS + c16,
                              dst + row * LDSTRIDE + c16);
            }
        }

        const float* bLds = Bs + buf * BS_FLOATS + l15 * LDSTRIDE + 2 * half;

        v8f c = {};
#pragma unroll
        for (int kk = 0; kk < DIMS; kk += 32) {      // 8 WMMAs per group
            v2f av[8], bv[8];
#pragma unroll
            for (int u = 0; u < 8; ++u) {
                av[u] = *(const v2f*)(aLds + kk + 4 * u);
                bv[u] = *(const v2f*)(bLds + kk + 4 * u);
            }
#pragma unroll
            for (int u = 0; u < 8; ++u)
                c = __builtin_amdgcn_wmma_f32_16x16x4_f32(
                        false, av[u], false, bv[u], (short)0, c, false, false);
        }

        const float invNj = invnorm[j0 + l15];
        const float fj    = flagf[j0 + l15];
        const int   j     = j0 + l15;
#pragma unroll
        for (int r = 0; r < 8; ++r) {
            const int i = i0 + r + 8 * half;
            float sim = c[r] * (invNi[r] * invNj * INV_T);
            float e   = __expf(sim);
            e = (i == j) ? 0.0f : e;          // zero the diagonal
            allSum[r] += e;
            posSum[r] += e * fj;              // positives: flagged columns
        }

        wait_async0();                        // next tile landed in LDS
        __syncthreads();                      // everyone done reading buf
        buf ^= 1;
    }

    // reduce over the 16 lanes sharing one m (n dimension), then atomics
#pragma unroll
    for (int r = 0; r < 8; ++r) {
        float a = allSum[r], p = posSum[r];
#pragma unroll
        for (int m = 8; m >= 1; m >>= 1) {
            a += __shfl_xor(a, m, 32);
            p += __shfl_xor(p, m, 32);
        }
        if (l15 == 0) {
            const int i = i0 + r + 8 * half;
            atomicAdd(&all_acc[i], a);
            atomicAdd(&pos_acc[i], p);
        }
    }
}

// ---------------------------------------------------------------------------
// Kernel 3: loss = mean over valid rows of  log(all_exp) - log(pos_exp)
// ---------------------------------------------------------------------------
__global__ void __launch_bounds__(1024)
finalize_kernel(const float* __restrict__ flagf,
                const float* __restrict__ all_acc,
                const float* __restrict__ pos_acc,
                float* __restrict__ out) {
    __shared__ float sA[32];
    __shared__ float sB[32];
    const int tid  = threadIdx.x;
    const int lane = tid & 31;
    const int wv   = tid >> 5;
    const int nthr = blockDim.x;

    float nf = 0.0f;
    for (int i = tid; i < NROWS; i += nthr) nf += flagf[i];
#pragma unroll
    for (int m = 16; m >= 1; m >>= 1) nf += __shfl_xor(nf, m, 32);
    if (lane == 0) sA[wv] = nf;
    __syncthreads();
    float nflag = 0.0f;
    for (int k = 0; k < 32; ++k) nflag += sA[k];
    __syncthreads();

    float lsum = 0.0f, cnt = 0.0f;
    if (nflag >= 2.0f) {
        for (int i = tid; i < NROWS; i += nthr) {
            if (flagf[i] != 0.0f) {
                lsum += logf(all_acc[i]) - logf(pos_acc[i]);
                cnt  += 1.0f;
            }
        }
    }
#pragma unroll
    for (int m = 16; m >= 1; m >>= 1) {
        lsum += __shfl_xor(lsum, m, 32);
        cnt  += __shfl_xor(cnt,  m, 32);
    }
    if (lane == 0) { sA[wv] = lsum; sB[wv] = cnt; }
    __syncthreads();
    if (tid == 0) {
        float L = 0.0f, C = 0.0f;
        for (int k = 0; k < 32; ++k) { L += sA[k]; C += sB[k]; }
        out[0] = (C > 0.0f) ? (L / C) : 0.0f;
    }
}

// ---------------------------------------------------------------------------
extern "C" void kernel_launch(void* const* d_in, const int* in_sizes, int n_in,
                              void* d_out, int out_size, void* d_ws, size_t ws_size,
                              hipStream_t stream) {
    (void)in_sizes; (void)n_in; (void)out_size; (void)ws_size;
    // inputs: 0=z_original (unused by reference), 1=z_flowed, 2=attributes
    const float* z    = (const float*)d_in[1];
    const float* attr = (const float*)d_in[2];
    float* out = (float*)d_out;

    float* ws       = (float*)d_ws;   // 4*NROWS floats = 128 KB scratch
    float* invnorm  = ws + 0 * NROWS;
    float* flagf    = ws + 1 * NROWS;
    float* all_acc  = ws + 2 * NROWS;
    float* pos_acc  = ws + 3 * NROWS;

    prep_kernel<<<NROWS / 8, 256, 0, stream>>>(z, attr, invnorm, flagf,
                                               all_acc, pos_acc);

    dim3 grid(NROWS / MBLOCK, JSLICES);
    gemm_exp_kernel<<<grid, 256, LDS_BYTES, stream>>>(z, invnorm, flagf,
                                                      all_acc, pos_acc);

    finalize_kernel<<<1, 1024, 0, stream>>>(flagf, all_acc, pos_acc, out);
}